// Loss_1065151889956
// MI455X (gfx1250) — compile-verified
//
#include <hip/hip_runtime.h>
#include <math.h>

typedef __attribute__((ext_vector_type(2))) float v2f;
typedef __attribute__((ext_vector_type(8))) float v8f;

#define ORIG      512
#define POOLED    256
#define OUTW      246          // 256 - 11 + 1
#define TILES_1D  16           // ceil(246/16)
#define NIMG      96           // 32 batch * 3 channels
#define NTILES    (NIMG * TILES_1D * TILES_1D)   // 24576
#define PSTRIDE   28           // pooled panel stride (conflict-free vs 64 banks)
#define PSIZE     (28 * 28)    // 784 floats per pooled field
#define HSTRIDE   17           // padded H panel stride (kills 2-way store conflicts)
#define HSIZE     (28 * 17)    // 476 floats
#define WAVE_LDS  (2 * PSIZE + HSIZE)  // 2044 floats per wave
#define WAVES_PER_BLOCK 4
#define WTAB      48           // zero-padded 11-tap table: taps at [16..26]

__global__ __launch_bounds__(128, 2)
void ssim_tile_kernel(const float* __restrict__ X, const float* __restrict__ Y,
                      float* __restrict__ partial)
{
    __shared__ float smem[WAVES_PER_BLOCK * WAVE_LDS + WTAB];   // 32,896 B

    const int wave = threadIdx.x >> 5;
    const int lane = threadIdx.x & 31;
    const int hi   = lane >> 4;     // half-wave select (K/M split per ISA layouts)
    const int ln   = lane & 15;

    float* Px   = &smem[wave * WAVE_LDS];
    float* Py   = Px + PSIZE;
    float* Hb   = Py + PSIZE;
    float* wtab = &smem[WAVES_PER_BLOCK * WAVE_LDS];

    // ---- Gaussian 1-D taps -> zero-padded LDS table (once per block) -------
    // Loaded (not computed) band operands cannot be rematerialized by the
    // compiler inside the WMMA loops, unlike the previous expf-based version.
    if (threadIdx.x == 0) {
        float g[11]; float s = 0.f;
#pragma unroll
        for (int i = 0; i < 11; ++i) {
            float d = (float)i - 5.0f;
            g[i] = expf(-(d * d) * (1.0f / 4.5f));
            s += g[i];
        }
        float inv = 1.0f / s;
#pragma unroll
        for (int i = 0; i < WTAB; ++i) wtab[i] = 0.0f;
#pragma unroll
        for (int i = 0; i < 11; ++i) wtab[16 + i] = g[i] * inv;
    }

    const int tile = blockIdx.x * WAVES_PER_BLOCK + wave;
    const int tx   = tile & 15;
    const int ty   = (tile >> 4) & 15;
    const int img  = tile >> 8;                 // b*3 + c

    const float* xi = X + (size_t)img * (ORIG * ORIG);
    const float* yi = Y + (size_t)img * (ORIG * ORIG);

    // ---- 2x2 average pool into LDS panels (clamped at image edge) ----------
    const int pr0 = ty * 16, pc0 = tx * 16;
    for (int idx = lane; idx < PSIZE; idx += 32) {
        int pr = idx / PSTRIDE, pc = idx - pr * PSTRIDE;
        int gr = pr0 + pr; if (gr > POOLED - 1) gr = POOLED - 1;
        int gc = pc0 + pc; if (gc > POOLED - 1) gc = POOLED - 1;
        size_t o0 = (size_t)(2 * gr)     * ORIG + (size_t)(2 * gc);
        size_t o1 = (size_t)(2 * gr + 1) * ORIG + (size_t)(2 * gc);
        float2 xa = *(const float2*)(xi + o0);
        float2 xb = *(const float2*)(xi + o1);
        float2 ya = *(const float2*)(yi + o0);
        float2 yb = *(const float2*)(yi + o1);
        Px[idx] = 0.25f * (xa.x + xa.y + xb.x + xb.y);
        Py[idx] = 0.25f * (ya.x + ya.y + yb.x + yb.y);
    }
    __syncthreads();

    // ---- Band operand, shared by hconv-B and vconv-A -----------------------
    // element (lane, VGPR v, chunk j) = w[K - n], K = 4j + v + 2*hi, n = lane&15
    // index 16 + K - n spans [1, 43] -> always in-table, no select needed.
    v2f Wb[7];
#pragma unroll
    for (int j = 0; j < 7; ++j) {
#pragma unroll
        for (int v = 0; v < 2; ++v) {
            Wb[j][v] = wtab[16 + 4 * j + v + 2 * hi - ln];
        }
    }

    // ---- 5 fields: separable 11x11 conv via V_WMMA_F32_16X16X4_F32 ---------
    v8f Vacc[5];
#pragma unroll
    for (int f = 0; f < 5; ++f) {
        // horizontal pass: H rows 0..27 in two 16-row groups (rbase 0 and 12)
#pragma unroll
        for (int g = 0; g < 2; ++g) {
            const int rbase = g ? 12 : 0;
            const int prow  = rbase + ln;        // A: M = lane&15
            v8f acc = {};
#pragma unroll
            for (int j = 0; j < 7; ++j) {
                v2f a;
#pragma unroll
                for (int v = 0; v < 2; ++v) {
                    int pcol = 4 * j + v + 2 * hi;   // A: K = v + 2*hi + 4j
                    float px = Px[prow * PSTRIDE + pcol];
                    float py = Py[prow * PSTRIDE + pcol];
                    a[v] = (f == 0) ? px
                         : (f == 1) ? py
                         : (f == 2) ? px * px
                         : (f == 3) ? py * py
                         :            px * py;
                }
                acc = __builtin_amdgcn_wmma_f32_16x16x4_f32(
                          false, a, false, Wb[j], (short)0, acc, false, false);
            }
            // C/D layout: VGPR r -> row (rbase + r + 8*hi), col = lane&15
#pragma unroll
            for (int r = 0; r < 8; ++r) {
                int row = rbase + r + 8 * hi;
                Hb[row * HSTRIDE + ln] = acc[r];
            }
        }
        __syncthreads();

        // vertical pass: V(16x16) = BandA(16x28) x H(28x16)
        v8f acc = {};
#pragma unroll
        for (int j = 0; j < 7; ++j) {
            v2f b;
#pragma unroll
            for (int v = 0; v < 2; ++v) {
                int hrow = 4 * j + v + 2 * hi;   // B: K-row = v + 2*hi (+4j)
                b[v] = Hb[hrow * HSTRIDE + ln];
            }
            acc = __builtin_amdgcn_wmma_f32_16x16x4_f32(
                      false, Wb[j], false, b, (short)0, acc, false, false);
        }
        Vacc[f] = acc;
        __syncthreads();   // protect Hb before next field overwrites it
    }

    // ---- SSIM map, masked partial sum --------------------------------------
    const float c1 = 1.0e-4f;   // 0.01^2
    const float c2 = 9.0e-4f;   // 0.03^2
    const int gcol = tx * 16 + ln;
    float sum = 0.f;
#pragma unroll
    for (int r = 0; r < 8; ++r) {
        int grow = ty * 16 + r + 8 * hi;
        float mux = Vacc[0][r], muy = Vacc[1][r];
        float mxx = mux * mux, myy = muy * muy, mxy = mux * muy;
        float sxx = Vacc[2][r] - mxx;
        float syy = Vacc[3][r] - myy;
        float sxy = Vacc[4][r] - mxy;
        float cs  = (2.0f * sxy + c2) / (sxx + syy + c2);
        float ss  = (2.0f * mxy + c1) / (mxx + myy + c1) * cs;
        if (grow < OUTW && gcol < OUTW) sum += ss;
    }
#pragma unroll
    for (int off = 16; off > 0; off >>= 1)
        sum += __shfl_down(sum, off, 32);
    if (lane == 0) partial[tile] = sum;
}

// ---- deterministic final reduction: 1 - mean(ssim) -------------------------
__global__ __launch_bounds__(256)
void ssim_reduce_kernel(const float* __restrict__ partial, float* __restrict__ out)
{
    __shared__ float s[256];
    float acc = 0.f;
    for (int i = threadIdx.x; i < NTILES; i += 256) acc += partial[i];
    s[threadIdx.x] = acc;
    __syncthreads();
#pragma unroll
    for (int k = 128; k > 0; k >>= 1) {
        if ((int)threadIdx.x < k) s[threadIdx.x] += s[threadIdx.x + k];
        __syncthreads();
    }
    if (threadIdx.x == 0)
        out[0] = 1.0f - s[0] * (1.0f / (32.0f * 3.0f * 246.0f * 246.0f));
}

extern "C" void kernel_launch(void* const* d_in, const int* in_sizes, int n_in,
                              void* d_out, int out_size, void* d_ws, size_t ws_size,
                              hipStream_t stream)
{
    const float* X = (const float*)d_in[0];
    const float* Y = (const float*)d_in[1];
    float* partial = (float*)d_ws;     // NTILES floats (96 KB), fully rewritten each call

    ssim_tile_kernel<<<NTILES / WAVES_PER_BLOCK, 128, 0, stream>>>(X, Y, partial);
    ssim_reduce_kernel<<<1, 256, 0, stream>>>(partial, (float*)d_out);
}